// GraphConv_82772609729053
// MI455X (gfx1250) — compile-verified
//
#include <hip/hip_runtime.h>
#include <hip/hip_bf16.h>

typedef __attribute__((ext_vector_type(2))) float v2f;
typedef __attribute__((ext_vector_type(8))) float v8f;

#define C_DIM 64
#define BN_EPS 1e-5f

// ---------------------------------------------------------------------------
// 0) zero fill
// ---------------------------------------------------------------------------
__global__ __launch_bounds__(256) void k_zero(float* __restrict__ p, size_t n) {
    size_t i = (size_t)blockIdx.x * blockDim.x + threadIdx.x;
    size_t stride = (size_t)gridDim.x * blockDim.x;
    for (; i < n; i += stride) p[i] = 0.0f;
}

// ---------------------------------------------------------------------------
// 1) Dual GEMM via V_WMMA_F32_16X16X4_F32, weights staged in LDS.
//    X   : [64][N]  (row-major, row stride N)
//    Wv, Wn : [64][64] row-major ([M][K])  -> staged to 32KB LDS per block
//    Fvt : [N][64]  node-major
//    Fnt : [N][64]  node-major
//    One wave per 16-column N-tile; 4 M-tiles x 2 weights = 8 accumulators.
// ---------------------------------------------------------------------------
__global__ __launch_bounds__(256) void k_gemm_wmma(
    const float* __restrict__ X,
    const float* __restrict__ Wv,
    const float* __restrict__ Wn,
    float* __restrict__ Fvt,
    float* __restrict__ Fnt,
    int N)
{
    __shared__ float lW[2 * C_DIM * C_DIM];       // [0..4095]=Wv, [4096..]=Wn

    // cooperative stage of both 64x64 weight matrices (float4-coalesced)
    {
        const float4* wv4 = reinterpret_cast<const float4*>(Wv);
        const float4* wn4 = reinterpret_cast<const float4*>(Wn);
        float4* l4 = reinterpret_cast<float4*>(lW);
#pragma unroll
        for (int i = 0; i < 4; ++i) {
            const int idx = threadIdx.x + i * 256;
            l4[idx]        = wv4[idx];
            l4[idx + 1024] = wn4[idx];
        }
    }
    __syncthreads();

    const int lane  = threadIdx.x & 31;
    const int wave  = threadIdx.x >> 5;
    const int ntile = blockIdx.x * 8 + wave;
    const long long n0 = (long long)ntile * 16;
    if (n0 >= N) return;                       // wave-uniform guard (post-barrier)

    const int row = lane & 15;                 // N-col (B/C/D) or M-row (A)
    const int hi  = lane >> 4;                 // lane-group select
    const float* lWv = lW;
    const float* lWn = lW + C_DIM * C_DIM;

    if (n0 + 16 <= N) {
        const long long n = n0 + row;
        v8f accV[4] = {v8f{}, v8f{}, v8f{}, v8f{}};
        v8f accN[4] = {v8f{}, v8f{}, v8f{}, v8f{}};

        for (int k0 = 0; k0 < C_DIM; k0 += 4) {
            const int kA = k0 + 2 * hi;        // K index for vgpr 0 of A/B
            v2f b;
            b.x = X[(size_t)(kA)     * N + n];
            b.y = X[(size_t)(kA + 1) * N + n];
#pragma unroll
            for (int mt = 0; mt < 4; ++mt) {
                const int m = mt * 16 + row;   // A: lane holds row M=lane&15
                v2f av, an;
                av.x = lWv[m * C_DIM + kA];
                av.y = lWv[m * C_DIM + kA + 1];
                an.x = lWn[m * C_DIM + kA];
                an.y = lWn[m * C_DIM + kA + 1];
                accV[mt] = __builtin_amdgcn_wmma_f32_16x16x4_f32(
                    false, av, false, b, (short)0, accV[mt], false, false);
                accN[mt] = __builtin_amdgcn_wmma_f32_16x16x4_f32(
                    false, an, false, b, (short)0, accN[mt], false, false);
            }
        }

        // C/D layout: VGPR r -> M = mt*16 + r + 8*hi, N-col = lane&15.
        // Node-major stores: 8 consecutive floats per (lane, mt) -> b128 pairs.
#pragma unroll
        for (int mt = 0; mt < 4; ++mt) {
#pragma unroll
            for (int r = 0; r < 8; ++r) {
                const int m = mt * 16 + r + 8 * hi;
                Fvt[(size_t)n * C_DIM + m] = accV[mt][r];
                Fnt[(size_t)n * C_DIM + m] = accN[mt][r];
            }
        }
    } else {
        // scalar tail (N not multiple of 16); never taken for N=100000
        const long long n = n0 + lane;
        if (n < N && lane < 16) {
            for (int m = 0; m < C_DIM; ++m) {
                float sv = 0.f, sn = 0.f;
                for (int k = 0; k < C_DIM; ++k) {
                    const float x = X[(size_t)k * N + n];
                    sv += lWv[m * C_DIM + k] * x;
                    sn += lWn[m * C_DIM + k] * x;
                }
                Fvt[(size_t)n * C_DIM + m] = sv;
                Fnt[(size_t)n * C_DIM + m] = sn;
            }
        }
    }
}

// ---------------------------------------------------------------------------
// 2) Edge scatter-add: 16 lanes per edge, float4 per lane (256B per edge).
// ---------------------------------------------------------------------------
__global__ __launch_bounds__(256) void k_edge_scatter(
    const float* __restrict__ Fnt,
    const int* __restrict__ gather_index,
    const int* __restrict__ reduce_index,
    float* __restrict__ sums,
    float* __restrict__ counts,
    int E)
{
    const int t = threadIdx.x & 15;
    const long long e = (long long)blockIdx.x * 16 + (threadIdx.x >> 4);
    if (e >= E) return;
    const int g = gather_index[e];
    const int r = reduce_index[e];
    const float4 v = *reinterpret_cast<const float4*>(
        Fnt + (size_t)g * C_DIM + t * 4);
    float* s = sums + (size_t)r * C_DIM + t * 4;
    atomicAdd(s + 0, v.x);
    atomicAdd(s + 1, v.y);
    atomicAdd(s + 2, v.z);
    atomicAdd(s + 3, v.w);
    if (t == 0) atomicAdd(counts + r, 1.0f);
}

// ---------------------------------------------------------------------------
// 3) pre = sums/max(cnt,1) + Fvt  (in place into sums, fully coalesced),
//    plus per-channel sum / sumsq partials.  Block = 4 nodes x 64 channels.
// ---------------------------------------------------------------------------
__global__ __launch_bounds__(256) void k_finalize(
    float* __restrict__ sums,            // [N][64], updated in place
    const float* __restrict__ counts,    // [N]
    const float* __restrict__ Fvt,       // [N][64]
    float* __restrict__ bnsum,
    float* __restrict__ bnsq,
    int N)
{
    __shared__ float s1[256];
    __shared__ float s2[256];
    const int t  = threadIdx.x;
    const int c  = t & 63;
    const long long n = (long long)blockIdx.x * 4 + (t >> 6);

    float val = 0.0f;
    if (n < N) {
        float cnt = counts[n];
        cnt = cnt > 1.0f ? cnt : 1.0f;
        const size_t i = (size_t)n * C_DIM + c;
        val = sums[i] / cnt + Fvt[i];
        sums[i] = val;
    }
    s1[t] = val;
    s2[t] = val * val;
    __syncthreads();
    if (t < 128) { s1[t] += s1[t + 128]; s2[t] += s2[t + 128]; }
    __syncthreads();
    if (t < 64) {
        atomicAdd(&bnsum[c], s1[t] + s1[t + 64]);
        atomicAdd(&bnsq[c],  s2[t] + s2[t + 64]);
    }
}

// ---------------------------------------------------------------------------
// 4) fold BN stats into per-channel scale/shift
// ---------------------------------------------------------------------------
__global__ void k_bn_params(
    const float* __restrict__ bnsum,
    const float* __restrict__ bnsq,
    const float* __restrict__ gamma,
    const float* __restrict__ beta,
    float* __restrict__ scale,
    float* __restrict__ shift,
    int N)
{
    const int c = threadIdx.x;
    if (c < C_DIM) {
        const float inv = 1.0f / (float)N;
        const float mu  = bnsum[c] * inv;
        const float var = bnsq[c] * inv - mu * mu;
        const float s   = gamma[c] * rsqrtf(var + BN_EPS);
        scale[c] = s;
        shift[c] = beta[c] - mu * s;
    }
}

// ---------------------------------------------------------------------------
// 5) BN + PReLU fused with node-major -> channel-major transpose through a
//    padded LDS tile; both global access patterns are 128B-coalesced.
//    Block = 32 nodes x 64 channels.
// ---------------------------------------------------------------------------
__global__ __launch_bounds__(256) void k_bn_prelu_tr(
    const float* __restrict__ pre,       // [N][64] (the updated sums buffer)
    float* __restrict__ out,             // [64][N]
    const float* __restrict__ scale,
    const float* __restrict__ shift,
    const float* __restrict__ prelu_a,
    int N)
{
    __shared__ float tile[32][C_DIM + 1];
    const int t = threadIdx.x;
    const long long n0 = (long long)blockIdx.x * 32;
    const float a = prelu_a[0];

#pragma unroll
    for (int i = 0; i < 8; ++i) {
        const int idx = t + i * 256;
        const int nl = idx >> 6;
        const int c  = idx & 63;
        const long long n = n0 + nl;
        tile[nl][c] = (n < N) ? pre[(size_t)n * C_DIM + c] : 0.0f;
    }
    __syncthreads();
#pragma unroll
    for (int i = 0; i < 8; ++i) {
        const int idx = t + i * 256;
        const int c  = idx >> 5;
        const int nl = idx & 31;
        const long long n = n0 + nl;
        if (n < N) {
            float v = tile[nl][c] * scale[c] + shift[c];
            out[(size_t)c * N + n] = v >= 0.0f ? v : a * v;
        }
    }
}

// ---------------------------------------------------------------------------
extern "C" void kernel_launch(void* const* d_in, const int* in_sizes, int n_in,
                              void* d_out, int out_size, void* d_ws, size_t ws_size,
                              hipStream_t stream)
{
    const float* X       = (const float*)d_in[0];  // [1,64,N]
    const float* Wv      = (const float*)d_in[1];  // [64,64]
    const float* Wn      = (const float*)d_in[2];  // [64,64]
    const float* gamma   = (const float*)d_in[3];  // [64]
    const float* beta    = (const float*)d_in[4];  // [64]
    const float* prelu_a = (const float*)d_in[5];  // [1]
    const int*   ridx    = (const int*)d_in[6];    // [E] reduce_index
    const int*   gidx    = (const int*)d_in[7];    // [E] gather_index

    const int N = in_sizes[0] / C_DIM;
    const int E = in_sizes[6];

    float* ws     = (float*)d_ws;
    float* Fvt    = ws;                            // N*64 (node-major)
    float* Fnt    = Fvt    + (size_t)C_DIM * N;    // N*64 (node-major)
    float* sums   = Fnt    + (size_t)C_DIM * N;    // N*64 (node-major)
    float* counts = sums   + (size_t)C_DIM * N;    // N
    float* bnsum  = counts + N;                    // 64
    float* bnsq   = bnsum  + C_DIM;                // 64
    float* scale  = bnsq   + C_DIM;                // 64
    float* shift  = scale  + C_DIM;                // 64

    // zero the accumulated regions: sums, counts, bnsum, bnsq
    const size_t zcount = (size_t)C_DIM * N + (size_t)N + 2 * C_DIM;
    k_zero<<<2048, 256, 0, stream>>>(sums, zcount);

    // dual GEMM (WMMA, LDS-staged weights): 8 N-tiles of 16 per block
    const int ntiles = (N + 15) / 16;
    k_gemm_wmma<<<(ntiles + 7) / 8, 256, 0, stream>>>(X, Wv, Wn, Fvt, Fnt, N);

    // edge gather + scatter-add (16 edges per block)
    k_edge_scatter<<<(E + 15) / 16, 256, 0, stream>>>(Fnt, gidx, ridx, sums,
                                                      counts, E);

    // mean + Fv -> pre-BN activations (in place), per-channel stats partials
    k_finalize<<<(N + 3) / 4, 256, 0, stream>>>(sums, counts, Fvt,
                                                bnsum, bnsq, N);

    // BN scale/shift
    k_bn_params<<<1, C_DIM, 0, stream>>>(bnsum, bnsq, gamma, beta, scale,
                                         shift, N);

    // BN apply + PReLU + transpose to [64][N]
    k_bn_prelu_tr<<<(N + 31) / 32, 256, 0, stream>>>(sums, (float*)d_out,
                                                     scale, shift, prelu_a, N);
}